// Sample_PDF_47588237639988
// MI455X (gfx1250) — compile-verified
//
#include <hip/hip_runtime.h>

// sample_pdf (mip-NeRF inverse-CDF sampling) for MI455X / gfx1250.
// R=65536 rays, N=64 coarse bins, 129 stratified u's -> [R,128,2] output.
// Bandwidth-bound (~145 MiB total traffic). The 64-wide per-ray prefix sum
// is computed with V_WMMA_F32_16X16X4_F32 (S = W x UpperTriOnes), 16 rays
// per wave, 40 WMMAs per wave.

#define NUM_RAYS 65536
#define NCOARSE  64
#define FINE     128   // output pairs per ray
#define ULEN     129

typedef float v2f __attribute__((ext_vector_type(2)));
typedef float v8f __attribute__((ext_vector_type(8)));

__device__ __forceinline__ float invert_cdf(const float* __restrict__ cdf,
                                            const float* __restrict__ tv,
                                            float u0)
{
    // searchsorted(cdf, u0, side='right'): cdf[0]=0 <= u0 < 1 = cdf[64],
    // so result idx is interior; track lo = idx-1, hi = idx.
    int lo = 0, hi = 64;
    float clo = 0.0f;   // cdf[0] written as exactly 0
    float chi = 1.0f;   // cdf[64] written as exactly 1
#pragma unroll
    for (int s = 0; s < 6; ++s) {          // 64 -> 1 interval width
        int   mid = (lo + hi) >> 1;
        float cm  = cdf[mid];
        bool  le  = (cm <= u0);
        lo  = le ? mid : lo;
        clo = le ? cm  : clo;
        hi  = le ? hi  : mid;
        chi = le ? chi : cm;
    }
    float denom = chi - clo;
    float t = (u0 - clo) / denom;
    if (!(t == t)) t = 0.0f;               // matches t[t != t] = 0
    t = fminf(fmaxf(t, 0.0f), 1.0f);
    float b0 = tv[lo], b1 = tv[hi];
    return b0 + t * (b1 - b0);
}

__global__ __launch_bounds__(128)
void sample_pdf_kernel(const float* __restrict__ weights,
                       const float* __restrict__ t_inters,
                       const float* __restrict__ u,
                       float* __restrict__ out)
{
    constexpr int WAVES = 4;   // waves per block
    constexpr int RPW   = 16;  // rays per wave (WMMA M dimension)
    constexpr int STR   = 68;  // padded LDS row stride (272 B, 16B aligned)

    __shared__ float ldsA[WAVES][RPW][STR]; // raw weights -> (reused) cdf[0..64]
    __shared__ float ldsB[WAVES][RPW][STR]; // blurred w   -> (reused) t_vals[0..64]

    const int tid     = threadIdx.x;
    const int wv      = tid >> 5;
    const int lane    = tid & 31;
    const int m       = lane & 15;   // WMMA row within A / column within D
    const int half    = lane >> 4;
    const int rayBase = blockIdx.x * (WAVES * RPW) + wv * RPW;

    // ---- phase 1: coalesced load of weights (16 rays x 64 f32) into LDS ----
    {
        const float4* wg = reinterpret_cast<const float4*>(
            weights + (size_t)rayBase * NCOARSE);
#pragma unroll
        for (int it = 0; it < 8; ++it) {
            int    q = it * 32 + lane;        // float4 index, 0..255
            float4 v = wg[q];
            int    r = q >> 4;                // ray within wave
            int    c = (q & 15) << 2;         // bin column
            *reinterpret_cast<float4*>(&ldsA[wv][r][c]) = v;
        }
    }
    __syncthreads();

    // ---- phase 2: maxblur + 0.01 floor -> ldsB ----
    // w[i] = 0.5*(max(w[i-1],w[i]) + max(w[i],w[i+1])) + 0.01, edges clamped.
    {
#pragma unroll
        for (int it = 0; it < 8; ++it) {
            int f0 = it * 128 + lane * 4;
            int r  = f0 >> 6;
            int c0 = f0 & 63;
            const float* row  = &ldsA[wv][r][0];
            float*       orow = &ldsB[wv][r][0];
#pragma unroll
            for (int j = 0; j < 4; ++j) {
                int   c   = c0 + j;
                float wm  = row[c == 0 ? 0 : c - 1];
                float wc  = row[c];
                float wp  = row[c == 63 ? 63 : c + 1];
                orow[c] = 0.5f * (fmaxf(wm, wc) + fmaxf(wc, wp)) + 0.01f;
            }
        }
    }
    __syncthreads();

    // ---- phase 3: inclusive cumsum via WMMA, then cdf -> ldsA ----
    // S[16x64] = Wblur[16x64] x U[64x64], U[j][k] = (j <= k).
    // A 16x4 f32 layout: lane (m, half): VGPR0 = K(2*half), VGPR1 = K(2*half+1).
    // B 4x16 layout (row-striped like C/D): VGPRv, lane: row K = v + 2*half,
    // col N = lane&15. D 16x16: component g = row (g + 8*half), col = lane&15.
    {
        v8f acc[4] = {};
        const float* wbm = &ldsB[wv][m][0];
        const int    nn  = m;                  // my D/B column
#pragma unroll
        for (int kb = 0; kb < 16; ++kb) {      // K blocks of 4
            const int kbase = kb * 4;
            v2f a;
            a.x = wbm[kbase + 2 * half];
            a.y = wbm[kbase + 2 * half + 1];
            const int qt = kb >> 2;            // diagonal output tile
#pragma unroll
            for (int nt = 0; nt < 4; ++nt) {
                if (nt < qt) continue;         // U block is zero
                v2f b;
                if (nt == qt) {                // triangular diagonal 16-block
                    int ncol = nt * 16 + nn;
                    b.x = (kbase + 2 * half     <= ncol) ? 1.0f : 0.0f;
                    b.y = (kbase + 2 * half + 1 <= ncol) ? 1.0f : 0.0f;
                } else {                       // strictly above diagonal: ones
                    b.x = 1.0f; b.y = 1.0f;
                }
                acc[nt] = __builtin_amdgcn_wmma_f32_16x16x4_f32(
                    false, a, false, b, (short)0, acc[nt], false, false);
            }
        }
        // cdf[k] = min(1, (S[k-1] + k*pad/64) / (wsum+pad)), cdf[0]=0, cdf[64]=1
#pragma unroll
        for (int g = 0; g < 8; ++g) {
            // S[row][63] lives at column lane 15 (rows 0-7) / 31 (rows 8-15)
            float ws   = __shfl(acc[3][g], (lane & 16) + 15, 32);
            float pad  = fmaxf(0.0f, 1e-5f - ws);
            float wsp  = ws + pad;
            float padq = pad * (1.0f / 64.0f);
            float* crow = &ldsA[wv][g + 8 * half][0];
#pragma unroll
            for (int nt = 0; nt < 4; ++nt) {
                int n = nt * 16 + nn;
                if (n <= 62) {
                    float s = acc[nt][g] + (float)(n + 1) * padq;
                    crow[n + 1] = fminf(1.0f, s / wsp);
                }
            }
        }
        if (half == 0) ldsA[wv][m][0]  = 0.0f;
        else           ldsA[wv][m][64] = 1.0f;
    }
    __syncthreads();

    // ---- phase 4: t_vals (interval starts + final end) -> ldsB ----
    {
        const float2* tg = reinterpret_cast<const float2*>(
            t_inters + (size_t)rayBase * (2 * NCOARSE));
#pragma unroll
        for (int it = 0; it < 32; ++it) {
            int    p = it * 32 + lane;        // float2 index, 0..1023
            float2 v = tg[p];
            int    r = p >> 6;
            int    i = p & 63;
            ldsB[wv][r][i] = v.x;             // interval start
            if (i == 63) ldsB[wv][r][64] = v.y; // last interval end
        }
    }
    __syncthreads();

    // ---- phase 5: inverse-CDF sampling; output [ray][f][2] ----
    // u strictly increasing + monotone inversion => samples already sorted,
    // so the reference's sort is the identity and is skipped.
    {
        float2* out2 = reinterpret_cast<float2*>(out);
        for (int r = 0; r < RPW; ++r) {
            const int    ray  = rayBase + r;
            const float* ug   = u + (size_t)ray * ULEN;
            const float* crow = &ldsA[wv][r][0];
            const float* trow = &ldsB[wv][r][0];
#pragma unroll
            for (int it = 0; it < 4; ++it) {
                int   f    = it * 32 + lane;
                float s_lo = invert_cdf(crow, trow, ug[f]);
                float s_hi = __shfl_down(s_lo, 1, 32);
                if (lane == 31) s_hi = invert_cdf(crow, trow, ug[f + 1]);
                out2[(size_t)ray * FINE + f] = make_float2(s_lo, s_hi);
            }
        }
    }
}

extern "C" void kernel_launch(void* const* d_in, const int* in_sizes, int n_in,
                              void* d_out, int out_size, void* d_ws, size_t ws_size,
                              hipStream_t stream)
{
    const float* weights  = (const float*)d_in[0]; // [R, 64]
    const float* t_inters = (const float*)d_in[1]; // [R, 64, 2]
    const float* u        = (const float*)d_in[2]; // [R, 129]
    float*       out      = (float*)d_out;         // [R, 128, 2]

    (void)in_sizes; (void)n_in; (void)out_size; (void)d_ws; (void)ws_size;

    dim3 block(128);                 // 4 waves, 16 rays/wave
    dim3 grid(NUM_RAYS / 64);        // 64 rays per block -> 1024 blocks
    sample_pdf_kernel<<<grid, block, 0, stream>>>(weights, t_inters, u, out);
}